// SpectralConv2d_80685255622856
// MI455X (gfx1250) — compile-verified
//
#include <hip/hip_runtime.h>
#include <hip/hip_bf16.h>

typedef __attribute__((ext_vector_type(2))) float v2f;
typedef __attribute__((ext_vector_type(8))) float v8f;

#define Z8 {0.f, 0.f, 0.f, 0.f, 0.f, 0.f, 0.f, 0.f}

namespace {

constexpr int Bsz  = 16;    // batch
constexpr int CIN  = 64;
constexpr int COUT = 64;
constexpr int Nn   = 128;   // H == W
constexpr int Mm   = 20;    // retained modes
constexpr int MP   = 32;    // padded modes
constexpr float W0 = 0.04908738521234052f; // 2*pi/128

// D = A(16x4) * B(4x16) + C, fp32 WMMA (V_WMMA_F32_16X16X4_F32)
__device__ __forceinline__ v8f wmma4(v2f a, v2f b, v8f c) {
  return __builtin_amdgcn_wmma_f32_16x16x4_f32(false, a, false, b, (short)0, c,
                                               false, false);
}

// Async global -> LDS copies (ISA 15.18.3, tracked by ASYNCcnt).
__device__ __forceinline__ void async_ld_b128(void* lds_dst, const void* gsrc) {
  asm volatile("global_load_async_to_lds_b128 %0, %1, off"
               :: "v"((unsigned)(uintptr_t)lds_dst),
                  "v"((unsigned long long)(uintptr_t)gsrc)
               : "memory");
}
__device__ __forceinline__ void async_ld_b32(void* lds_dst, const void* gsrc) {
  asm volatile("global_load_async_to_lds_b32 %0, %1, off"
               :: "v"((unsigned)(uintptr_t)lds_dst),
                  "v"((unsigned long long)(uintptr_t)gsrc)
               : "memory");
}
__device__ __forceinline__ void wait_async0() {
  asm volatile("s_wait_asynccnt 0x0" ::: "memory");
}

// ---------------------------------------------------------------------------
// Basis construction.
// DHT kernel cos(a+b) - sin(a'+b') splits into 4 separable terms:
//   (+ C1*C2) (- S1*S2) (- S1'*C2') (- C1'*S2'),  primes = index+1 phases.
// Forward:  U = x * Bcat   (Bcat[n2][32t+k2] = col bases, zero for k2>=20)
//           xm = A2^T * Ut (A2[t*128+n1][k1]  = signed row bases)
// Inverse:  Wt[32t+k1][p2] = z * Vcat (Vcat[k2][128t+p2])
//           y  = Ainv^T * Wt (Ainv[32t+k1][p1], scaled by 1/(H*W))
// ---------------------------------------------------------------------------
__global__ __launch_bounds__(256) void build_basis(float* __restrict__ Bcat,
                                                   float* __restrict__ A2,
                                                   float* __restrict__ Vcat,
                                                   float* __restrict__ Ainv) {
  int id = blockIdx.x * 256 + threadIdx.x;  // 0 .. 65535
  if (id < 16384) {                         // Bcat: 128 x 128
    int n2 = id >> 7, c = id & 127;
    int t = c >> 5, k2 = c & 31;
    float v = 0.0f;
    if (k2 < Mm) {
      int nn = (t >= 2) ? n2 + 1 : n2;
      float ph = W0 * (float)(nn * k2);
      v = (t & 1) ? sinf(ph) : cosf(ph);
    }
    Bcat[id] = v;
  } else if (id < 32768) {                  // A2: 512 x 32
    int j = id - 16384;
    int r = j >> 5, k1 = j & 31;
    int t = r >> 7, n1 = r & 127;
    float v = 0.0f;
    if (k1 < Mm) {
      int nn = (t >= 2) ? n1 + 1 : n1;
      float ph = W0 * (float)(nn * k1);
      float cv = cosf(ph), sv = sinf(ph);
      v = (t == 0) ? cv : ((t == 3) ? -cv : -sv);
    }
    A2[j] = v;
  } else if (id < 49152) {                  // Vcat: 32 x 512
    int j = id - 32768;
    int k2 = j >> 9, c = j & 511;
    int t = c >> 7, p2 = c & 127;
    int kk = (t >= 2) ? k2 + 1 : k2;
    float ph = W0 * (float)(kk * p2);
    Vcat[j] = (t & 1) ? sinf(ph) : cosf(ph);
  } else {                                  // Ainv: 128 x 128 (w/ 1/16384)
    int j = id - 49152;
    int r = j >> 7, p1 = j & 127;
    int t = r >> 5, k1 = r & 31;
    int kk = (t >= 2) ? k1 + 1 : k1;
    float ph = W0 * (float)(kk * p1);
    float cv = cosf(ph), sv = sinf(ph);
    float v = (t == 0) ? cv : ((t == 3) ? -cv : -sv);
    Ainv[j] = v * (1.0f / 16384.0f);
  }
}

// wp = 0.5*(w1 + neg_k(w1)), wm = 0.5*(w1 - neg_k(w1)); layout [mode][i][o]
__global__ __launch_bounds__(256) void prep_w(const float* __restrict__ w1,
                                              float* __restrict__ wp,
                                              float* __restrict__ wm) {
  int j = blockIdx.x * 256 + threadIdx.x;   // 400*4096 = 1,638,400
  int mode = j >> 12;
  int io = j & 4095;
  int i = io >> 6, o = io & 63;
  int k1 = mode / Mm, k2 = mode % Mm;
  int k1n = (Mm - k1) % Mm, k2n = (Mm - k2) % Mm;
  float a = w1[((i * COUT + o) * Mm + k1) * Mm + k2];
  float b = w1[((i * COUT + o) * Mm + k1n) * Mm + k2n];
  wp[j] = 0.5f * (a + b);
  wm[j] = 0.5f * (a - b);
}

__global__ void zero_buf(float* __restrict__ p, int n) {
  int i = blockIdx.x * 256 + threadIdx.x;
  if (i < n) p[i] = 0.0f;
}

// ---------------------------------------------------------------------------
// Forward truncated DHT per (b,c) image: xm[img][32][32] (pads naturally 0)
// ---------------------------------------------------------------------------
__global__ __launch_bounds__(256) void fwd_dht(const float* __restrict__ x,
                                               const float* __restrict__ Bcat,
                                               const float* __restrict__ A2,
                                               float* __restrict__ xm) {
  __shared__ float xs[128][132];  // padded: A-operand row-per-lane reads
  __shared__ float Us[512][33];   // Ut[(t*128+n1)][k2]
  const int img = blockIdx.x;     // 0..1023 == b*64+cin
  const float* xi = x + (size_t)img * (Nn * Nn);

  // Async-stage the 128x128 image into LDS (16B per lane per issue).
  for (int i4 = threadIdx.x; i4 < (Nn * Nn) / 4; i4 += 256) {
    int r = i4 >> 5, c = (i4 & 31) << 2;
    async_ld_b128(&xs[r][c], xi + (size_t)i4 * 4);
  }
  wait_async0();
  __syncthreads();

  const int lane = threadIdx.x & 31;
  const int m = lane & 15;
  const int hi = lane >> 4;
  const int w = threadIdx.x >> 5;  // wave 0..7

  // GEMM1: U(128x128) = xs * Bcat ; scattered into Us[(t,n1)][k2]
  for (int tc = 0; tc < 8; ++tc) {
    v8f acc = Z8;
    const int col = tc * 16 + m;
    for (int k = 0; k < 128; k += 4) {
      v2f a, b;
      a.x = xs[w * 16 + m][k + 2 * hi];
      a.y = xs[w * 16 + m][k + 2 * hi + 1];
      b.x = Bcat[(k + 2 * hi) * 128 + col];
      b.y = Bcat[(k + 2 * hi + 1) * 128 + col];
      acc = wmma4(a, b, acc);
    }
    const int t = col >> 5, k2 = col & 31;
#pragma unroll
    for (int vv = 0; vv < 8; ++vv) {
      int n1 = w * 16 + vv + 8 * hi;
      Us[t * 128 + n1][k2] = acc[vv];
    }
  }
  __syncthreads();

  // GEMM2: xm(32x32) = A2^T(32x512) * Us(512x32); one 16x16 tile per wave 0..3
  if (w < 4) {
    const int tr = w >> 1, tcb = w & 1;
    v8f acc = Z8;
    for (int k = 0; k < 512; k += 4) {
      v2f a, b;
      a.x = A2[(k + 2 * hi) * 32 + tr * 16 + m];
      a.y = A2[(k + 2 * hi + 1) * 32 + tr * 16 + m];
      b.x = Us[k + 2 * hi][tcb * 16 + m];
      b.y = Us[k + 2 * hi + 1][tcb * 16 + m];
      acc = wmma4(a, b, acc);
    }
    float* xo = xm + (size_t)img * (MP * MP);
#pragma unroll
    for (int vv = 0; vv < 8; ++vv)
      xo[(tr * 16 + vv + 8 * hi) * MP + tcb * 16 + m] = acc[vv];
  }
}

// ---------------------------------------------------------------------------
// Spectral channel mix per mode: out[b,o] = X*wp + Xn*wm  (K=128 fused GEMM)
// ---------------------------------------------------------------------------
__global__ __launch_bounds__(128) void spectral(const float* __restrict__ xm,
                                                const float* __restrict__ wp,
                                                const float* __restrict__ wm,
                                                float* __restrict__ ym) {
  __shared__ float As[16][132];          // cols 0..63 = X, 64..127 = Xn
  const int mode = blockIdx.x;           // 0..399
  const int k1 = mode / Mm, k2 = mode % Mm;
  const int k1n = (Mm - k1) % Mm, k2n = (Mm - k2) % Mm;
  const int md = k1 * MP + k2;
  const int mdn = k1n * MP + k2n;

  // Async-gather the strided A panel straight into LDS.
  for (int i = threadIdx.x; i < 16 * 128; i += 128) {
    int b = i >> 7, c = i & 127, ci = c & 63;
    async_ld_b32(&As[b][c],
                 &xm[(size_t)(b * CIN + ci) * (MP * MP) + ((c < 64) ? md : mdn)]);
  }
  wait_async0();
  __syncthreads();

  const int lane = threadIdx.x & 31, m = lane & 15, hi = lane >> 4;
  const int w = threadIdx.x >> 5;        // Cout block 0..3
  const float* wpm = wp + (size_t)mode * (CIN * COUT);
  const float* wmm = wm + (size_t)mode * (CIN * COUT);
  const int n = w * 16 + m;
  v8f acc = Z8;
  for (int k = 0; k < 128; k += 4) {
    int kk0 = k + 2 * hi;                // even; whole step on one side of 64
    v2f a, b;
    a.x = As[m][kk0];
    a.y = As[m][kk0 + 1];
    b.x = (kk0 < 64) ? wpm[kk0 * 64 + n] : wmm[(kk0 - 64) * 64 + n];
    b.y = (kk0 < 64) ? wpm[(kk0 + 1) * 64 + n] : wmm[(kk0 - 63) * 64 + n];
    acc = wmma4(a, b, acc);
  }
#pragma unroll
  for (int vv = 0; vv < 8; ++vv) {
    int bb = vv + 8 * hi;
    ym[(size_t)(bb * COUT + n) * (MP * MP) + md] = acc[vv];
  }
}

// ---------------------------------------------------------------------------
// Inverse DHT per (b,o) image: y[128][128] = Ainv^T * (z * Vcat)
// ---------------------------------------------------------------------------
__global__ __launch_bounds__(256) void inv_dht(const float* __restrict__ ym,
                                               const float* __restrict__ Vcat,
                                               const float* __restrict__ Ainv,
                                               float* __restrict__ y) {
  __shared__ float z[32][36];
  __shared__ float Wt[128][132];
  const int img = blockIdx.x;            // 0..1023 == b*64+cout
  const float* zi = ym + (size_t)img * (MP * MP);
  {                                      // 256 lanes x 16B == whole 4KB block
    int i4 = threadIdx.x;
    int r = i4 >> 3, c = (i4 & 7) << 2;
    async_ld_b128(&z[r][c], zi + (size_t)i4 * 4);
  }
  wait_async0();
  __syncthreads();

  const int lane = threadIdx.x & 31, m = lane & 15, hi = lane >> 4;
  const int w = threadIdx.x >> 5;
  const int t = w >> 1, k1b = (w & 1) * 16;

  // Stage 1: Wt[(32t+k1)][p2] = sum_k2 z[k1][k2] * Vcat[k2][128t+p2]
  for (int ct = 0; ct < 8; ++ct) {
    v8f acc = Z8;
    for (int k = 0; k < 32; k += 4) {
      v2f a, b;
      a.x = z[k1b + m][k + 2 * hi];
      a.y = z[k1b + m][k + 2 * hi + 1];
      b.x = Vcat[(k + 2 * hi) * 512 + t * 128 + ct * 16 + m];
      b.y = Vcat[(k + 2 * hi + 1) * 512 + t * 128 + ct * 16 + m];
      acc = wmma4(a, b, acc);
    }
#pragma unroll
    for (int vv = 0; vv < 8; ++vv)
      Wt[w * 16 + vv + 8 * hi][ct * 16 + m] = acc[vv];
  }
  __syncthreads();

  // Stage 2: y[p1][p2] = sum_r Ainv[r][p1] * Wt[r][p2]
  float* yo = y + (size_t)img * (Nn * Nn);
  for (int ct = 0; ct < 8; ++ct) {
    v8f acc = Z8;
    for (int k = 0; k < 128; k += 4) {
      v2f a, b;
      a.x = Ainv[(k + 2 * hi) * 128 + w * 16 + m];
      a.y = Ainv[(k + 2 * hi + 1) * 128 + w * 16 + m];
      b.x = Wt[k + 2 * hi][ct * 16 + m];
      b.y = Wt[k + 2 * hi + 1][ct * 16 + m];
      acc = wmma4(a, b, acc);
    }
#pragma unroll
    for (int vv = 0; vv < 8; ++vv)
      yo[(w * 16 + vv + 8 * hi) * Nn + ct * 16 + m] = acc[vv];
  }
}

// ---------------------------------------------------------------------------
// 9-tap Gaussian blur along one power-of-2 axis (edge clamp), optional W mask.
// ss = log2(stride), ls = log2(len). Memory bound: L2-resident at 192MB.
// ---------------------------------------------------------------------------
__global__ __launch_bounds__(256) void blur_axis(const float* __restrict__ in,
                                                 float* __restrict__ out,
                                                 int ss, int ls, int apply_mask) {
  const int idx = blockIdx.x * 256 + threadIdx.x;   // 2^24 elements
  const int len = 1 << ls;
  const int inner = idx & ((1 << ss) - 1);
  const int tmp = idx >> ss;
  const int pos = tmp & (len - 1);
  const int outer = tmp >> ls;
  const size_t base = ((size_t)outer << (ss + ls)) + (size_t)inner;

  const float e1 = 0.60653065971263342f;   // exp(-0.5)
  const float e2 = 0.13533528323661270f;   // exp(-2.0)
  const float e3 = 0.011108996538242306f;  // exp(-4.5)
  const float e4 = 0.00033546262790251185f;// exp(-8.0)
  const float inv = 1.0f / (1.0f + 2.0f * (e1 + e2 + e3 + e4));
  const float g[9] = {e4 * inv, e3 * inv, e2 * inv, e1 * inv, inv,
                      e1 * inv, e2 * inv, e3 * inv, e4 * inv};

  float acc = 0.0f;
#pragma unroll
  for (int t = 0; t < 9; ++t) {
    int p = pos + t - 4;
    p = p < 0 ? 0 : (p > len - 1 ? len - 1 : p);
    acc += g[t] * in[base + ((size_t)p << ss)];
  }
  if (apply_mask && (pos > 32 && pos < 96)) acc = 0.0f;  // |fftfreq| > 0.25
  out[idx] = acc;
}

}  // namespace

extern "C" void kernel_launch(void* const* d_in, const int* in_sizes, int n_in,
                              void* d_out, int out_size, void* d_ws, size_t ws_size,
                              hipStream_t stream) {
  (void)in_sizes; (void)n_in; (void)out_size; (void)ws_size;
  const float* x  = (const float*)d_in[0];   // (16,64,128,128)
  const float* w1 = (const float*)d_in[1];   // (64,64,20,20)
  float* out = (float*)d_out;                // (16,64,128,128)
  float* ws  = (float*)d_ws;

  // Small region (lifetime: phases 1-6), 5.44M floats:
  float* Bcat = ws;                  // 16384
  float* A2   = Bcat + 16384;        // 16384
  float* Vcat = A2 + 16384;          // 16384
  float* Ainv = Vcat + 16384;        // 16384
  float* wp   = Ainv + 16384;        // 1,638,400
  float* wm   = wp + 1638400;        // 1,638,400
  float* xmb  = wm + 1638400;        // 1,048,576
  float* ymb  = xmb + 1048576;       // 1,048,576
  // Blur ping-pong buffer (lifetime: phases 7-10) aliases the small region:
  float* ybuf = ws;                  // 16,777,216 floats (64 MiB)

  build_basis<<<256, 256, 0, stream>>>(Bcat, A2, Vcat, Ainv);
  prep_w<<<6400, 256, 0, stream>>>(w1, wp, wm);
  zero_buf<<<4096, 256, 0, stream>>>(ymb, 1048576);
  fwd_dht<<<1024, 256, 0, stream>>>(x, Bcat, A2, xmb);
  spectral<<<400, 128, 0, stream>>>(xmb, wp, wm, ymb);
  inv_dht<<<1024, 256, 0, stream>>>(ymb, Vcat, Ainv, out);
  // Separable 4-D Gaussian: out -> ybuf -> out -> ybuf -> out (ends in d_out)
  blur_axis<<<65536, 256, 0, stream>>>(out, ybuf, 20, 4, 0);  // axis B
  blur_axis<<<65536, 256, 0, stream>>>(ybuf, out, 14, 6, 0);  // axis C
  blur_axis<<<65536, 256, 0, stream>>>(out, ybuf, 7, 7, 0);   // axis H
  blur_axis<<<65536, 256, 0, stream>>>(ybuf, out, 0, 7, 1);   // axis W + mask
}